// MvdrBeamformer_31971736551855
// MI455X (gfx1250) — compile-verified
//
#include <hip/hip_runtime.h>

typedef __attribute__((ext_vector_type(2))) float v2f;
typedef __attribute__((ext_vector_type(8))) float v8f;

#define NB 8
#define NM 8
#define NF 512
#define NT 1000

// ---------------------------------------------------------------------------
// Kernel 1: per-(b,f) MVDR weight computation.
//   - principal eigenvector of target SCM via power iteration (Hermitian PSD)
//   - complex Cholesky solve R_n x = d
//   - w = x / (d^H x); store [Re(w)[0..7], Im(w)[0..7]] per (b,f) in wbuf
// One thread per (b,f); f is the fastest dim of the SCM layout so SCM loads
// are coalesced across consecutive threads.
// ---------------------------------------------------------------------------
__global__ __launch_bounds__(256) void mvdr_weights_kernel(
    const float* __restrict__ tsr, const float* __restrict__ tsi,
    const float* __restrict__ nsr, const float* __restrict__ nsi,
    float* __restrict__ wbuf)
{
  int idx = blockIdx.x * blockDim.x + threadIdx.x;
  if (idx >= NB * NF) return;
  const int b = idx / NF;
  const int f = idx % NF;

  float Ar[NM][NM], Ai[NM][NM];

  // ---- load target SCM (B, M, M, F) ----
  #pragma unroll
  for (int i = 0; i < NM; ++i) {
    #pragma unroll
    for (int j = 0; j < NM; ++j) {
      const int off = ((b * NM + i) * NM + j) * NF + f;
      Ar[i][j] = tsr[off];
      Ai[i][j] = tsi[off];
    }
  }

  // ---- power iteration for principal eigenvector d ----
  float vr[NM], vi[NM];
  #pragma unroll
  for (int i = 0; i < NM; ++i) { vr[i] = Ar[i][0]; vi[i] = Ai[i][0]; }

  for (int it = 0; it < 40; ++it) {
    float yr[NM], yi[NM];
    float nrm = 0.0f;
    #pragma unroll
    for (int i = 0; i < NM; ++i) {
      float sr = 0.0f, si = 0.0f;
      #pragma unroll
      for (int j = 0; j < NM; ++j) {
        sr += Ar[i][j] * vr[j] - Ai[i][j] * vi[j];
        si += Ar[i][j] * vi[j] + Ai[i][j] * vr[j];
      }
      yr[i] = sr; yi[i] = si;
      nrm += sr * sr + si * si;
    }
    const float s = rsqrtf(fmaxf(nrm, 1e-30f));
    #pragma unroll
    for (int i = 0; i < NM; ++i) { vr[i] = yr[i] * s; vi[i] = yi[i] * s; }
  }

  // ---- load noise SCM into same arrays, complex Cholesky N = L L^H ----
  #pragma unroll
  for (int i = 0; i < NM; ++i) {
    #pragma unroll
    for (int j = 0; j < NM; ++j) {
      const int off = ((b * NM + i) * NM + j) * NF + f;
      Ar[i][j] = nsr[off];
      Ai[i][j] = nsi[off];
    }
  }

  #pragma unroll
  for (int k = 0; k < NM; ++k) {
    float diag = Ar[k][k];
    #pragma unroll
    for (int j = 0; j < NM; ++j) {
      if (j < k) diag -= Ar[k][j] * Ar[k][j] + Ai[k][j] * Ai[k][j];
    }
    diag = sqrtf(fmaxf(diag, 1e-20f));
    Ar[k][k] = diag; Ai[k][k] = 0.0f;
    const float inv = 1.0f / diag;
    #pragma unroll
    for (int i = 0; i < NM; ++i) {
      if (i > k) {
        float sr = Ar[i][k], si = Ai[i][k];
        #pragma unroll
        for (int j = 0; j < NM; ++j) {
          if (j < k) {
            // L[i][j] * conj(L[k][j])
            sr -= Ar[i][j] * Ar[k][j] + Ai[i][j] * Ai[k][j];
            si -= Ai[i][j] * Ar[k][j] - Ar[i][j] * Ai[k][j];
          }
        }
        Ar[i][k] = sr * inv; Ai[i][k] = si * inv;
      }
    }
  }

  // ---- forward solve L y = d ----
  float yr2[NM], yi2[NM];
  #pragma unroll
  for (int i = 0; i < NM; ++i) {
    float sr = vr[i], si = vi[i];
    #pragma unroll
    for (int j = 0; j < NM; ++j) {
      if (j < i) {
        // L[i][j] * y[j]
        sr -= Ar[i][j] * yr2[j] - Ai[i][j] * yi2[j];
        si -= Ar[i][j] * yi2[j] + Ai[i][j] * yr2[j];
      }
    }
    const float inv = 1.0f / Ar[i][i];
    yr2[i] = sr * inv; yi2[i] = si * inv;
  }

  // ---- back solve L^H x = y ----
  float xr[NM], xi[NM];
  #pragma unroll
  for (int ii = 0; ii < NM; ++ii) {
    const int i = NM - 1 - ii;
    float sr = yr2[i], si = yi2[i];
    #pragma unroll
    for (int j = 0; j < NM; ++j) {
      if (j > i) {
        // conj(L[j][i]) * x[j]
        sr -= Ar[j][i] * xr[j] + Ai[j][i] * xi[j];
        si -= Ar[j][i] * xi[j] - Ai[j][i] * xr[j];
      }
    }
    const float inv = 1.0f / Ar[i][i];
    xr[i] = sr * inv; xi[i] = si * inv;
  }

  // ---- den = Re(d^H x); w = x / den ----
  float den = 0.0f;
  #pragma unroll
  for (int m = 0; m < NM; ++m) den += vr[m] * xr[m] + vi[m] * xi[m];
  const float invden = 1.0f / den;

  float* wo = wbuf + idx * 16;
  #pragma unroll
  for (int m = 0; m < NM; ++m) {
    wo[m]     = xr[m] * invden;  // Re(w)
    wo[8 + m] = xi[m] * invden;  // Im(w)
  }
}

// ---------------------------------------------------------------------------
// Kernel 2: beamforming apply via V_WMMA_F32_16X16X4_F32.
//   y(t) = sum_m conj(w[m]) x[m,t]
// Real formulation with K = 16 stacked rows [xr(0..7); xi(0..7)]:
//   y_re row coefficients: [ wr , wi ]   (A-matrix row M=0)
//   y_im row coefficients: [-wi , wr ]   (A-matrix row M=1)
// 4 chained K=4 WMMAs per 16-time tile. One block per (b,f); 8 waves split
// the time tiles. Pure HBM-bandwidth problem (~294 MB moved -> ~13 us at
// 23.3 TB/s); WMMA pipe is nowhere near a bottleneck at 2/16 M-utilization.
// ---------------------------------------------------------------------------
__global__ __launch_bounds__(256) void mvdr_apply_kernel(
    const float* __restrict__ mix_re,
    const float* __restrict__ mix_im,
    const float* __restrict__ wbuf,
    float* __restrict__ out)
{
  const int bf   = blockIdx.x;        // 0..B*F-1
  const int b    = bf / NF;
  const int f    = bf % NF;
  const int lane = threadIdx.x & 31;
  const int wave = threadIdx.x >> 5;  // 0..7

  const float* w = wbuf + bf * 16;    // [wr(0..7), wi(0..7)]

  const int mrow  = lane & 15;        // output row index (M)
  const int khalf = lane >> 4;        // 0: K base 0, 1: K base 2 (+2)

  // Build the 4 A-fragments once (weights constant over t).
  // 32-bit A 16x4 layout: VGPR j, lanes 0-15 -> K=j, lanes 16-31 -> K=j+2.
  v2f A[4];
  #pragma unroll
  for (int c = 0; c < 4; ++c) {
    #pragma unroll
    for (int j = 0; j < 2; ++j) {
      const int r = c * 4 + j + 2 * khalf;   // stacked row 0..15
      float val = 0.0f;
      if (mrow == 0) {
        val = w[r];                           // [wr, wi]
      } else if (mrow == 1) {
        val = (r < 8) ? -w[r + 8] : w[r - 8]; // [-wi, wr]
      }
      A[c][j] = val;
    }
  }

  const int ntiles = (NT + 15) / 16;   // 63
  const int tn = lane & 15;            // time index within tile (N)

  for (int tile = wave; tile < ntiles; tile += 8) {
    const int t0 = tile * 16;
    const int tt = t0 + tn;
    const int tl = tt < NT ? tt : NT - 1;   // clamp loads on last tile

    // prefetch next tile for this wave
    if (tile + 8 < ntiles) {
      __builtin_prefetch(&mix_re[((b * NM) * NF + f) * NT + t0 + 128], 0, 1);
      __builtin_prefetch(&mix_im[((b * NM) * NF + f) * NT + t0 + 128], 0, 1);
    }

    v8f acc = {};
    #pragma unroll
    for (int c = 0; c < 4; ++c) {
      // B 4x16 layout mirrors A: VGPR j, lanes 0-15 -> K=j, lanes16-31 -> K=j+2
      v2f Bf;
      #pragma unroll
      for (int j = 0; j < 2; ++j) {
        const int r = c * 4 + j + 2 * khalf;     // stacked row 0..15
        const int m = r & 7;
        const float* p = (r < 8) ? mix_re : mix_im;
        Bf[j] = p[((b * NM + m) * NF + f) * NT + tl];
      }
      acc = __builtin_amdgcn_wmma_f32_16x16x4_f32(
          /*neg_a=*/false, A[c], /*neg_b=*/false, Bf,
          /*c_mod=*/(short)0, acc, /*reuse_a=*/false, /*reuse_b=*/false);
    }

    // D layout: VGPR0 lanes 0-15 = (M=0, N=lane) -> y_re
    //           VGPR1 lanes 0-15 = (M=1, N=lane) -> y_im
    if (lane < 16 && tt < NT) {
      float2 o;
      o.x = acc[0];
      o.y = acc[1];
      *reinterpret_cast<float2*>(out + (size_t)(bf * NT + tt) * 2) = o;
    }
  }
}

// ---------------------------------------------------------------------------
extern "C" void kernel_launch(void* const* d_in, const int* in_sizes, int n_in,
                              void* d_out, int out_size, void* d_ws, size_t ws_size,
                              hipStream_t stream) {
  const float* mix_re = (const float*)d_in[0];
  const float* mix_im = (const float*)d_in[1];
  const float* tsr    = (const float*)d_in[2];
  const float* tsi    = (const float*)d_in[3];
  const float* nsr    = (const float*)d_in[4];
  const float* nsi    = (const float*)d_in[5];
  float* out  = (float*)d_out;   // complex64 (B,F,T) -> interleaved float pairs
  float* wbuf = (float*)d_ws;    // 4096 * 16 floats = 256 KB of scratch

  mvdr_weights_kernel<<<(NB * NF + 255) / 256, 256, 0, stream>>>(
      tsr, tsi, nsr, nsi, wbuf);
  mvdr_apply_kernel<<<NB * NF, 256, 0, stream>>>(mix_re, mix_im, wbuf, out);
}